// Gate_82463372083939
// MI455X (gfx1250) — compile-verified
//
#include <hip/hip_runtime.h>
#include <hip/hip_bf16.h>

typedef float v2f __attribute__((ext_vector_type(2)));
typedef float v8f __attribute__((ext_vector_type(8)));

#define NTOK        65536
#define DIM         4096
#define NEXP        64
#define TOPK_K      8
#define ROUTE_SCALE 2.5f

#define WAVES_PER_BLOCK 8
#define TOK_PER_WAVE    16
#define TOK_PER_BLOCK   (WAVES_PER_BLOCK * TOK_PER_WAVE)  // 128
#define LDS_STRIDE      68                                // 68*4 = 272 B, 16B aligned rows

__global__ __launch_bounds__(256, 2)
void moe_gate_wmma_kernel(const float* __restrict__ x,
                          const float* __restrict__ W,
                          float* __restrict__ out_w,
                          int*   __restrict__ out_i) {
  __shared__ float slds[WAVES_PER_BLOCK][TOK_PER_WAVE][LDS_STRIDE];

  const int lane = threadIdx.x & 31;
  const int wave = threadIdx.x >> 5;
  const int tok0 = blockIdx.x * TOK_PER_BLOCK + wave * TOK_PER_WAVE;

  const int halfw = lane >> 4;   // 0: lanes 0-15 (K=0,1), 1: lanes 16-31 (K=2,3)
  const int l16   = lane & 15;

  // A fragment: token row (tok0 + l16), columns k + 2*halfw + {0,1}  (float2)
  const float* arow = x + (size_t)(tok0 + l16) * DIM + 2 * halfw;
  // B fragment: expert row (16*t + l16), columns k + 2*halfw + {0,1} (float2)
  const float* brow = W + (size_t)l16 * DIM + 2 * halfw;

  v8f acc0 = {}, acc1 = {}, acc2 = {}, acc3 = {};

  for (int k = 0; k < DIM; k += 4) {
    // x streamed once -> non-temporal; W stays L2-resident -> regular loads
    v2f a  = __builtin_nontemporal_load((const v2f*)(arow + k));
    v2f b0 = *(const v2f*)(brow + (size_t)0  * DIM + k);
    v2f b1 = *(const v2f*)(brow + (size_t)16 * DIM + k);
    v2f b2 = *(const v2f*)(brow + (size_t)32 * DIM + k);
    v2f b3 = *(const v2f*)(brow + (size_t)48 * DIM + k);
    acc0 = __builtin_amdgcn_wmma_f32_16x16x4_f32(false, a, false, b0, (short)0, acc0, false, false);
    acc1 = __builtin_amdgcn_wmma_f32_16x16x4_f32(false, a, false, b1, (short)0, acc1, false, false);
    acc2 = __builtin_amdgcn_wmma_f32_16x16x4_f32(false, a, false, b2, (short)0, acc2, false, false);
    acc3 = __builtin_amdgcn_wmma_f32_16x16x4_f32(false, a, false, b3, (short)0, acc3, false, false);
  }

  // D layout: VGPR j, lanes 0-15 -> (M=j, N=lane), lanes 16-31 -> (M=j+8, N=lane-16)
  // token = j + 8*halfw, expert = 16*t + l16
  #pragma unroll
  for (int j = 0; j < 8; ++j) {
    const int tok = j + 8 * halfw;
    slds[wave][tok][ 0 + l16] = acc0[j];
    slds[wave][tok][16 + l16] = acc1[j];
    slds[wave][tok][32 + l16] = acc2[j];
    slds[wave][tok][48 + l16] = acc3[j];
  }
  __syncthreads();

  // Softmax + top-8 per token; lanes 0-15 each own one token.
  if (lane < 16) {
    float v[NEXP];
    #pragma unroll
    for (int i = 0; i < NEXP; i += 4) {
      const float4 q = *(const float4*)&slds[wave][lane][i];
      v[i + 0] = q.x; v[i + 1] = q.y; v[i + 2] = q.z; v[i + 3] = q.w;
    }

    float mx = v[0];
    #pragma unroll
    for (int i = 1; i < NEXP; ++i) mx = fmaxf(mx, v[i]);

    float denom = 0.0f;
    #pragma unroll
    for (int i = 0; i < NEXP; ++i) denom += __expf(v[i] - mx);
    const float inv = 1.0f / denom;

    float* wout = out_w + (size_t)(tok0 + lane) * TOPK_K;
    int*   iout = out_i + (size_t)(tok0 + lane) * TOPK_K;

    #pragma unroll
    for (int kk = 0; kk < TOPK_K; ++kk) {
      // argmax keeping first occurrence on ties (matches jax.lax.top_k)
      float m = v[0]; int mi = 0;
      #pragma unroll
      for (int i = 1; i < NEXP; ++i) {
        const bool g = v[i] > m;
        m  = g ? v[i] : m;
        mi = g ? i    : mi;
      }
      wout[kk] = __expf(m - mx) * inv * ROUTE_SCALE;
      iout[kk] = mi;
      // knock out the selected expert (no dynamic register indexing)
      #pragma unroll
      for (int i = 0; i < NEXP; ++i) {
        v[i] = (i == mi) ? -3.402823466e38f : v[i];
      }
    }
  }
}

extern "C" void kernel_launch(void* const* d_in, const int* in_sizes, int n_in,
                              void* d_out, int out_size, void* d_ws, size_t ws_size,
                              hipStream_t stream) {
  (void)in_sizes; (void)n_in; (void)out_size; (void)d_ws; (void)ws_size;
  const float* x = (const float*)d_in[0];
  const float* W = (const float*)d_in[1];
  float* out_w = (float*)d_out;                                   // 65536*8 fp32 weights
  int*   out_i = (int*)((float*)d_out + (size_t)NTOK * TOPK_K);   // 65536*8 int32 indices

  dim3 grid(NTOK / TOK_PER_BLOCK);  // 512 blocks
  dim3 block(256);                  // 8 waves of 32
  moe_gate_wmma_kernel<<<grid, block, 0, stream>>>(x, W, out_w, out_i);
}